// interactionModule_nonParametric_2Dacceleration_83614423319055
// MI455X (gfx1250) — compile-verified
//
#include <hip/hip_runtime.h>
#include <cstdint>

#define INV_2PI 0.15915494309189535f

// Abramowitz & Stegun K1 approximation, branch-free select (matches the
// jnp.where in the reference: both branches evaluated, then selected).
__device__ __forceinline__ float bessel_k1f(float x) {
    float xs = fmaxf(x, 1e-12f);
    // ---- small branch: 0 < x <= 2 ----
    float a  = xs * (1.0f / 3.75f);
    float t2 = a * a;
    float i1 = xs * (0.5f + t2 * (0.87890594f + t2 * (0.51498869f + t2 * (0.15084934f
             + t2 * (0.02658733f + t2 * (0.00301532f + t2 * 0.00032411f))))));
    float h = xs * 0.5f;
    float t = h * h;
    float poly = 0.15443144f + t * (-0.67278579f + t * (-0.18156897f + t * (-0.01919402f
               + t * (-0.00110404f + t * (-0.00004686f)))));
    float xk1   = xs * __logf(h) * i1 + 1.0f + t * poly;
    float small = xk1 / xs;
    // ---- large branch: x >= 2 ----
    float s  = 2.0f / xs;
    float lp = 1.25331414f + s * (0.23498619f + s * (-0.03655620f + s * (0.01504268f
             + s * (-0.00780353f + s * (0.00325614f + s * (-0.00068245f))))));
    float large = __expf(-xs) * rsqrtf(xs) * lp;
    return (xs <= 2.0f) ? small : large;
}

// Kernel 1: pack per-node data for single-gather access in the edge kernel:
//   nodeA[i] = (x, y, cos(theta), sin(theta))      -- one 16B line hit per endpoint
//   nodeB[i] = (A_CFs[celltype], A_chems[celltype]) -- dst-side coefficients
// and write the base output terms (u0*p, A_ext*cos(theta)).
__global__ void node_init_kernel(const float2* __restrict__ xpos,
                                 const float* __restrict__ theta,
                                 const int*   __restrict__ celltype,
                                 const float* __restrict__ u0p,
                                 const float* __restrict__ aextp,
                                 const float* __restrict__ acfs,
                                 const float* __restrict__ achems,
                                 float4* __restrict__ nodeA,
                                 float2* __restrict__ nodeB,
                                 float*  __restrict__ out,
                                 int n) {
    int i = blockIdx.x * blockDim.x + threadIdx.x;
    if (i >= n) return;
    float u0 = *u0p, aext = *aextp;
    float sn, cs;
    __sincosf(theta[i], &sn, &cs);
    float2 xi = xpos[i];
    nodeA[i] = make_float4(xi.x, xi.y, cs, sn);
    int ct = celltype[i];
    nodeB[i] = make_float2(ct ? acfs[1] : acfs[0], ct ? achems[1] : achems[0]);
    out[3 * i + 0] = u0 * cs;
    out[3 * i + 1] = u0 * sn;
    out[3 * i + 2] = aext * cs;
}

// Kernel 2: per-edge messages, scattered into out[dst] with HW f32 atomics.
// Each wave (32 lanes, wave32) stages 128 src + 128 dst indices per pass into
// its private LDS slab with async global->LDS DMA, tagged non-temporal: the
// 25.6 MB edge stream is read exactly once and must not evict the 2.4 MB
// packed node tables from WGP$/L2.
__global__ void __launch_bounds__(256)
edge_kernel(const float4* __restrict__ nodeA,
            const float2* __restrict__ nodeB,
            const int*    __restrict__ src,
            const int*    __restrict__ dst,
            const float*  __restrict__ kappap,
            const float*  __restrict__ rp,
            const float*  __restrict__ betap,
            const float*  __restrict__ acilp,
            float* __restrict__ out,
            int E) {
    __shared__ int smem[8 * 256];  // 8 waves x (128 src + 128 dst) ints = 8 KB

    const float kappa = *kappap;
    const float r     = *rp;
    const float beta  = *betap;
    const float A_CIL = *acilp;
    const float k1cut      = bessel_k1f(0.5f);   // CUTOFF = 0.5
    const float chem_scale = kappa * INV_2PI;

    const int wave = threadIdx.x >> 5;
    const int lane = threadIdx.x & 31;
    const int waveBase = blockIdx.x * 1024 + wave * 128;   // 128 edges per wave
    int* s_src = &smem[wave * 256];
    int* s_dst = s_src + 128;

    int4 sv, dv;
    const bool full = (waveBase + 128 <= E);
    if (full) {
        // LDS byte addresses (generic-pointer low 32 bits == DS address space)
        unsigned lsrc = (unsigned)(uintptr_t)(const void*)(s_src + lane * 4);
        unsigned ldst = (unsigned)(uintptr_t)(const void*)(s_dst + lane * 4);
        unsigned goff = (unsigned)(waveBase + lane * 4) * 4u;  // byte offset into edge arrays
        asm volatile(
            "global_load_async_to_lds_b128 %0, %2, %3 th:TH_LOAD_NT\n\t"
            "global_load_async_to_lds_b128 %1, %2, %4 th:TH_LOAD_NT"
            :
            : "v"(lsrc), "v"(ldst), "v"(goff), "s"(src), "s"(dst)
            : "memory");
        asm volatile("s_wait_asynccnt 0x0" ::: "memory");
        sv = *reinterpret_cast<const int4*>(s_src + lane * 4);
        dv = *reinterpret_cast<const int4*>(s_dst + lane * 4);
    }

#pragma unroll
    for (int j = 0; j < 4; ++j) {
        int e = waveBase + lane * 4 + j;
        int si, di;
        if (full) {
            si = (&sv.x)[j];
            di = (&dv.x)[j];
        } else {
            if (e >= E) continue;
            si = src[e];
            di = dst[e];
        }

        float4 ns = nodeA[si];   // (x, y, cos, sin) -- one 16B gather per endpoint
        float4 nd = nodeA[di];
        float2 ad = nodeB[di];   // (A_CF, A_chem) for dst celltype

        float dx = nd.x - ns.x;
        float dy = nd.y - ns.y;
        float absdr = sqrtf(dx * dx + dy * dy);
        float inv = 1.0f / fmaxf(absdr, 1e-12f);      // F.normalize clamp
        float ux = dx * inv, uy = dy * inv;

        float cross = uy * nd.z - ux * nd.w;          // p_dst x unit_dr
        float J_CIL = r / absdr - 1.0f;               // unclamped, as in reference
        float J_CF  = 0.5f * (1.0f + ux * ns.z + uy * ns.w);
        float J_chem = fmaxf(bessel_k1f(kappa * absdr) - k1cut, 0.0f) * chem_scale;

        float nbJ = -beta * J_CIL;
        float mvx = nbJ * ux;
        float mvy = nbJ * uy;
        float mt  = (ad.x * J_CF - A_CIL * J_CIL + ad.y * J_chem) * cross;

        float* o = out + 3 * di;
        unsafeAtomicAdd(o + 0, mvx);   // global_atomic_add_f32, L2-resolved
        unsafeAtomicAdd(o + 1, mvy);
        unsafeAtomicAdd(o + 2, mt);
    }
}

extern "C" void kernel_launch(void* const* d_in, const int* in_sizes, int n_in,
                              void* d_out, int out_size, void* d_ws, size_t ws_size,
                              hipStream_t stream) {
    const float2* x        = (const float2*)d_in[0];
    const float*  theta    = (const float*)d_in[1];
    const int*    celltype = (const int*)d_in[2];
    const int*    src      = (const int*)d_in[3];
    const int*    dst      = (const int*)d_in[4];
    const float*  kappa    = (const float*)d_in[5];
    const float*  r        = (const float*)d_in[6];
    const float*  u0       = (const float*)d_in[7];
    const float*  beta     = (const float*)d_in[8];
    const float*  A_CIL    = (const float*)d_in[9];
    const float*  A_ext    = (const float*)d_in[10];
    const float*  A_CFs    = (const float*)d_in[11];
    const float*  A_chems  = (const float*)d_in[12];

    int n = in_sizes[0] / 2;   // N_NODES
    int E = in_sizes[3];       // N_EDGES

    float4* nodeA = (float4*)d_ws;                      // N * 16 B
    float2* nodeB = (float2*)((char*)d_ws + (size_t)n * sizeof(float4));  // N * 8 B
    float* out = (float*)d_out;                         // [N,3] flat

    node_init_kernel<<<(n + 255) / 256, 256, 0, stream>>>(
        x, theta, celltype, u0, A_ext, A_CFs, A_chems, nodeA, nodeB, out, n);

    int nBlocks = (E + 1023) / 1024;   // 1024 edges per 256-thread block
    edge_kernel<<<nBlocks, 256, 0, stream>>>(
        nodeA, nodeB, src, dst, kappa, r, beta, A_CIL, out, E);
}